// MeSFormerBlock_56521769615662
// MI455X (gfx1250) — compile-verified
//
#include <hip/hip_runtime.h>
#include <hip/hip_bf16.h>

// ---------------------------------------------------------------------------
// MeSFormer block for MI455X (gfx1250, wave32).
// All GEMMs use V_WMMA_F32_16X16X4_F32 (native f32 matrix path).
// Fusion: xnorm(energy) is a per-row linear scale, so attention is single-pass:
//   O = diag(gamma * rsqrt(sqrt(sum E^4))) * (E @ Vn),  E never hits HBM.
// ---------------------------------------------------------------------------

typedef float v2f __attribute__((ext_vector_type(2)));
typedef float v8f __attribute__((ext_vector_type(8)));

#define B_ 2
#define S_ 2048
#define E_ 1024
#define H_ 16
#define D_ 64
#define GAMMA 0.01f
#define LN_EPS 1e-5f

// One 16x16x4 f32 WMMA step reading A (row-major, stride sa) and B
// (k-major, stride sb) from LDS.
//   A layout: lane m=lane&15 holds K = k0 + 2*(lane>>4) .. +1   (2 VGPRs)
//   B layout: lane n=lane&15 holds same K pair (2 VGPRs)
//   C/D layout: VGPR r: lanes0-15 -> M=r, lanes16-31 -> M=r+8, N=lane&15
__device__ __forceinline__ v8f wmma4(const float* __restrict__ As, int sa,
                                     const float* __restrict__ Bs, int sb,
                                     int k0, v8f c) {
  const int lane = threadIdx.x & 31;
  const int mn = lane & 15;
  const int kk = k0 + ((lane >> 4) << 1);
  v2f a, b;
  a.x = As[mn * sa + kk];
  a.y = As[mn * sa + kk + 1];
  b.x = Bs[kk * sb + mn];
  b.y = Bs[(kk + 1) * sb + mn];
  return __builtin_amdgcn_wmma_f32_16x16x4_f32(false, a, false, b,
                                               (short)0, c, false, false);
}

__device__ __forceinline__ v8f vzero8() {
  v8f z = {0.f, 0.f, 0.f, 0.f, 0.f, 0.f, 0.f, 0.f};
  return z;
}

// ---------------------------------------------------------------------------
// Kernel 1: per-head QKV projection, y = x @ W^T  (W is 64x64, shared by all
// heads).  grid = (B*S*H/16, 3), block = 128 (4 waves).  Each block handles
// 16 flattened (n,s,h) rows (== one (n,s) across all 16 heads).
// Output layout head-major: [n, h, s, d]  (contiguous [S,D] per (n,h)).
// For matrix 2 (V) the x-norm over d is fused.
// ---------------------------------------------------------------------------
__global__ __launch_bounds__(128) void k_qkv(
    const float* __restrict__ q_in, const float* __restrict__ k_in,
    const float* __restrict__ v_in, const float* __restrict__ Wq,
    const float* __restrict__ Wk, const float* __restrict__ Wv,
    float* __restrict__ qh, float* __restrict__ kh, float* __restrict__ vh) {
  __shared__ float Xs[16 * 68];   // A: 16 rows x 64 (pad 4)
  __shared__ float WT[64 * 68];   // B: WT[d][e] = W[e][d]
  __shared__ float Ys[16 * 68];   // staged result
  __shared__ float scl[16];

  const int tid = threadIdx.x;
  const int wave = tid >> 5;
  const int lane = tid & 31;
  const int mtx = blockIdx.y;  // 0=q,1=k,2=v

  const float* X = (mtx == 0) ? q_in : (mtx == 1) ? k_in : v_in;
  const float* W = (mtx == 0) ? Wq : (mtx == 1) ? Wk : Wv;
  float* dst = (mtx == 0) ? qh : (mtx == 1) ? kh : vh;

  const int row0 = blockIdx.x * 16;  // flattened (n*S+s)*H + h

  for (int i = tid; i < 16 * 64; i += 128) {
    Xs[(i >> 6) * 68 + (i & 63)] = X[row0 * 64 + i];
  }
  for (int i = tid; i < 64 * 64; i += 128) {
    const int e = i >> 6, d = i & 63;
    WT[d * 68 + e] = W[i];  // transpose into k-major
  }
  __syncthreads();

  v8f acc = vzero8();
  const int colbase = wave * 16;
#pragma unroll
  for (int k0 = 0; k0 < 64; k0 += 4)
    acc = wmma4(Xs, 68, WT + colbase, 68, k0, acc);

  // stage D tile to LDS (C/D layout -> row-major)
  const int n = lane & 15;
  const int mh = (lane >> 4) * 8;
#pragma unroll
  for (int r = 0; r < 8; ++r) Ys[(mh + r) * 68 + colbase + n] = acc[r];
  __syncthreads();

  if (mtx == 2) {  // fused 4-norm over d for V
    if (tid < 16) {
      float s4 = 0.f;
      for (int d = 0; d < 64; ++d) {
        float t = Ys[tid * 68 + d];
        t = t * t;
        s4 += t * t;
      }
      scl[tid] = GAMMA * __frsqrt_rn(__fsqrt_rn(s4));
    }
    __syncthreads();
  }

  for (int i = tid; i < 16 * 64; i += 128) {
    const int m = i >> 6, d = i & 63;
    const int row = row0 + m;
    const int nb = row >> 15;          // / (S*H)
    const int s = (row >> 4) & 2047;   // / H % S
    const int h = row & 15;
    float v = Ys[m * 68 + d];
    if (mtx == 2) v *= scl[m];
    dst[(((size_t)nb * H_ + h) * S_ + s) * D_ + d] = v;
  }
}

// ---------------------------------------------------------------------------
// Kernel 2: fused attention.  For each (n,h) and 64-row Q block:
//   loop over 128 K-chunks of 16: E16x16 = Q*K^T (WMMA), s4 += E^4,
//   stage E via LDS, O += E @ Vn (WMMA).  Final per-row scale
//   gamma * rsqrt(sqrt(s4)).  grid = (S/64, B*H), block = 128 (4 waves).
// ---------------------------------------------------------------------------
__global__ __launch_bounds__(128) void k_attn(
    const float* __restrict__ qh, const float* __restrict__ kh,
    const float* __restrict__ vh, float* __restrict__ attn) {
  __shared__ float Qs[64 * 68];        // 64 q rows x 64 d
  __shared__ float KsT[64 * 20];       // K chunk, d-major: KsT[d][ki]
  __shared__ float Vs[16 * 68];        // V chunk, ki-major: Vs[ki][d]
  __shared__ float Es[4][16 * 20];     // per-wave E staging (A layout feed)
  __shared__ float S4[4][32 * 8];      // per-lane s4 partials
  __shared__ float Scl[64];

  const int tid = threadIdx.x;
  const int wave = tid >> 5;
  const int lane = tid & 31;
  const int bh = blockIdx.y;           // n*H + h
  const int q0 = blockIdx.x * 64;

  const float* Q = qh + (size_t)bh * S_ * D_;
  const float* K = kh + (size_t)bh * S_ * D_;
  const float* V = vh + (size_t)bh * S_ * D_;

  for (int i = tid; i < 64 * 64; i += 128)
    Qs[(i >> 6) * 68 + (i & 63)] = Q[(size_t)(q0 + (i >> 6)) * D_ + (i & 63)];

  v8f oacc[4];
#pragma unroll
  for (int t = 0; t < 4; ++t) oacc[t] = vzero8();
  v8f s4p = vzero8();

  const int n = lane & 15;
  const int mh = (lane >> 4) * 8;

  for (int c = 0; c < S_ / 16; ++c) {
    __syncthreads();
    // stage K chunk (transposed to d-major) and V chunk
    for (int i = tid; i < 16 * 64; i += 128) {
      const int ki = i >> 6, d = i & 63;
      const float kv = K[(size_t)(c * 16 + ki) * D_ + d];
      KsT[d * 20 + ki] = kv;
      Vs[ki * 68 + d] = V[(size_t)(c * 16 + ki) * D_ + d];
    }
    if (c + 1 < S_ / 16) {  // prefetch next chunk -> global_prefetch_b8
      __builtin_prefetch(&K[(size_t)((c + 1) * 16) * D_ + tid * 8], 0, 1);
      __builtin_prefetch(&V[(size_t)((c + 1) * 16) * D_ + tid * 8], 0, 1);
    }
    __syncthreads();

    // E = Qtile @ Kchunk^T   (16x16, K=64 -> 16 WMMAs)
    v8f e = vzero8();
#pragma unroll
    for (int k0 = 0; k0 < 64; k0 += 4)
      e = wmma4(Qs + (wave * 16) * 68, 68, KsT, 20, k0, e);

    // s4 += E^4 ; stage E into A-readable layout
#pragma unroll
    for (int r = 0; r < 8; ++r) {
      float t = e[r] * e[r];
      s4p[r] += t * t;
      Es[wave][(mh + r) * 20 + n] = e[r];
    }
    // wave-local LDS RAW: LDS ops from one wave are serviced in order;
    // make the store->load ordering explicit.
    asm volatile("s_wait_dscnt 0" ::: "memory");

    // O += E @ Vn   (16x64, K=16 -> 16 WMMAs)
#pragma unroll
    for (int dt = 0; dt < 4; ++dt) {
#pragma unroll
      for (int k0 = 0; k0 < 16; k0 += 4)
        oacc[dt] = wmma4(Es[wave], 20, Vs + dt * 16, 68, k0, oacc[dt]);
    }
  }

  // reduce s4 across lanes -> per-row scale
#pragma unroll
  for (int r = 0; r < 8; ++r) S4[wave][lane * 8 + r] = s4p[r];
  __syncthreads();
  if (tid < 64) {
    const int w = tid >> 4, m = tid & 15;
    const int lbase = (m >> 3) * 16, r = m & 7;
    float s = 0.f;
    for (int l = 0; l < 16; ++l) s += S4[w][(lbase + l) * 8 + r];
    Scl[tid] = GAMMA * __frsqrt_rn(__fsqrt_rn(s));
  }
  __syncthreads();

  const int nb = bh >> 4;
  const int h = bh & 15;
#pragma unroll
  for (int dt = 0; dt < 4; ++dt) {
#pragma unroll
    for (int r = 0; r < 8; ++r) {
      const int m = mh + r;
      const int srow = q0 + wave * 16 + m;
      attn[((size_t)(nb * S_ + srow)) * E_ + h * D_ + dt * 16 + n] =
          oacc[dt][r] * Scl[wave * 16 + m];
    }
  }
}

// ---------------------------------------------------------------------------
// Kernel 3: x = LayerNorm(attn @ Wo^T + bo + query).  grid = B*S/16,
// block = 256 (8 waves); each wave owns a 128-col slab, K chunked by 8
// (Wo stays resident in the 192MB L2 across the 256 row-blocks).
// ---------------------------------------------------------------------------
__global__ __launch_bounds__(256) void k_wo_ln(
    const float* __restrict__ attn, const float* __restrict__ query,
    const float* __restrict__ Wo, const float* __restrict__ bo,
    const float* __restrict__ lnw, const float* __restrict__ lnb,
    float* __restrict__ xln) {
  __shared__ float As[16 * 12];        // 16 rows x 8 k (pad)
  __shared__ float Bs[8 * 1028];       // k-major Wo^T chunk
  __shared__ float Red[256 * 16];      // per-thread {sum[8], sumsq[8]}
  __shared__ float MuRs[16 * 2];

  const int tid = threadIdx.x;
  const int wave = tid >> 5;
  const int lane = tid & 31;
  const int row0 = blockIdx.x * 16;

  v8f acc[8];
#pragma unroll
  for (int t = 0; t < 8; ++t) acc[t] = vzero8();

  for (int kc = 0; kc < E_; kc += 8) {
    __syncthreads();
    if (tid < 128) {
      const int r = tid >> 3, kk = tid & 7;
      As[r * 12 + kk] = attn[(size_t)(row0 + r) * E_ + kc + kk];
    }
    for (int i = tid; i < 8 * 1024; i += 256) {
      const int kk = i & 7, nn = i >> 3;
      Bs[kk * 1028 + nn] = Wo[(size_t)nn * E_ + kc + kk];
    }
    __syncthreads();
#pragma unroll
    for (int dt = 0; dt < 8; ++dt) {
      const int colb = wave * 128 + dt * 16;
      acc[dt] = wmma4(As, 12, Bs + colb, 1028, 0, acc[dt]);
      acc[dt] = wmma4(As, 12, Bs + colb, 1028, 4, acc[dt]);
    }
  }

  // epilogue: + bo + query, then LayerNorm over 1024 cols
  const int n = lane & 15;
  const int mh = (lane >> 4) * 8;
  float psum[8], psq[8];
#pragma unroll
  for (int r = 0; r < 8; ++r) { psum[r] = 0.f; psq[r] = 0.f; }
#pragma unroll
  for (int dt = 0; dt < 8; ++dt) {
    const int col = wave * 128 + dt * 16 + n;
    const float bb = bo[col];
#pragma unroll
    for (int r = 0; r < 8; ++r) {
      const int m = mh + r;
      float v = acc[dt][r] + bb + query[(size_t)(row0 + m) * E_ + col];
      acc[dt][r] = v;
      psum[r] += v;
      psq[r] += v * v;
    }
  }
#pragma unroll
  for (int r = 0; r < 8; ++r) {
    Red[tid * 16 + r] = psum[r];
    Red[tid * 16 + 8 + r] = psq[r];
  }
  __syncthreads();
  if (tid < 16) {  // row m = tid
    const int half = tid >> 3, r = tid & 7;
    float s = 0.f, sq = 0.f;
    for (int w = 0; w < 8; ++w)
      for (int l = 0; l < 16; ++l) {
        const int t = w * 32 + half * 16 + l;
        s += Red[t * 16 + r];
        sq += Red[t * 16 + 8 + r];
      }
    const float mu = s * (1.f / E_);
    const float var = sq * (1.f / E_) - mu * mu;
    MuRs[tid * 2] = mu;
    MuRs[tid * 2 + 1] = __frsqrt_rn(var + LN_EPS);
  }
  __syncthreads();
#pragma unroll
  for (int dt = 0; dt < 8; ++dt) {
    const int col = wave * 128 + dt * 16 + n;
    const float g = lnw[col], bta = lnb[col];
#pragma unroll
    for (int r = 0; r < 8; ++r) {
      const int m = mh + r;
      const float v = (acc[dt][r] - MuRs[m * 2]) * MuRs[m * 2 + 1] * g + bta;
      xln[(size_t)(row0 + m) * E_ + col] = v;
    }
  }
}

// ---------------------------------------------------------------------------
// Kernel 4: out = x * sigmoid(x @ Wsq^T + bsq).  Same GEMM skeleton as K3.
// ---------------------------------------------------------------------------
__global__ __launch_bounds__(256) void k_se(
    const float* __restrict__ xln, const float* __restrict__ Wsq,
    const float* __restrict__ bsq, float* __restrict__ out) {
  __shared__ float As[16 * 12];
  __shared__ float Bs[8 * 1028];

  const int tid = threadIdx.x;
  const int wave = tid >> 5;
  const int lane = tid & 31;
  const int row0 = blockIdx.x * 16;

  v8f acc[8];
#pragma unroll
  for (int t = 0; t < 8; ++t) acc[t] = vzero8();

  for (int kc = 0; kc < E_; kc += 8) {
    __syncthreads();
    if (tid < 128) {
      const int r = tid >> 3, kk = tid & 7;
      As[r * 12 + kk] = xln[(size_t)(row0 + r) * E_ + kc + kk];
    }
    for (int i = tid; i < 8 * 1024; i += 256) {
      const int kk = i & 7, nn = i >> 3;
      Bs[kk * 1028 + nn] = Wsq[(size_t)nn * E_ + kc + kk];
    }
    __syncthreads();
#pragma unroll
    for (int dt = 0; dt < 8; ++dt) {
      const int colb = wave * 128 + dt * 16;
      acc[dt] = wmma4(As, 12, Bs + colb, 1028, 0, acc[dt]);
      acc[dt] = wmma4(As, 12, Bs + colb, 1028, 4, acc[dt]);
    }
  }

  const int n = lane & 15;
  const int mh = (lane >> 4) * 8;
#pragma unroll
  for (int dt = 0; dt < 8; ++dt) {
    const int col = wave * 128 + dt * 16 + n;
    const float bb = bsq[col];
#pragma unroll
    for (int r = 0; r < 8; ++r) {
      const int m = mh + r;
      const float y = acc[dt][r] + bb;
      const float x = xln[(size_t)(row0 + m) * E_ + col];
      const float gate = 1.f / (1.f + __expf(-y));
      out[(size_t)(row0 + m) * E_ + col] = x * gate;
    }
  }
}

// ---------------------------------------------------------------------------
extern "C" void kernel_launch(void* const* d_in, const int* in_sizes, int n_in,
                              void* d_out, int out_size, void* d_ws,
                              size_t ws_size, hipStream_t stream) {
  const float* value = (const float*)d_in[0];
  const float* key_i = (const float*)d_in[1];
  const float* query = (const float*)d_in[2];
  const float* Wq = (const float*)d_in[3];
  const float* Wk = (const float*)d_in[4];
  const float* Wv = (const float*)d_in[5];
  const float* Wo = (const float*)d_in[6];
  const float* bo = (const float*)d_in[7];
  const float* lnw = (const float*)d_in[8];
  const float* lnb = (const float*)d_in[9];
  const float* Wsq = (const float*)d_in[10];
  const float* bsq = (const float*)d_in[11];

  // workspace: qh | kh | vh | attn | xln   (5 x 16 MB f32)
  const size_t NTOK = (size_t)B_ * S_ * E_;  // 4,194,304
  float* ws = (float*)d_ws;
  float* qh = ws;
  float* kh = qh + NTOK;
  float* vh = kh + NTOK;
  float* attn = vh + NTOK;
  float* xln = attn + NTOK;

  k_qkv<<<dim3((B_ * S_ * H_) / 16, 3), 128, 0, stream>>>(
      query, key_i, value, Wq, Wk, Wv, qh, kh, vh);
  k_attn<<<dim3(S_ / 64, B_ * H_), 128, 0, stream>>>(qh, kh, vh, attn);
  k_wo_ln<<<(B_ * S_) / 16, 256, 0, stream>>>(attn, query, Wo, bo, lnw, lnb,
                                              xln);
  k_se<<<(B_ * S_) / 16, 256, 0, stream>>>(xln, Wsq, bsq, (float*)d_out);
}